// EntNet_13400297963637
// MI455X (gfx1250) — compile-verified
//
#include <hip/hip_runtime.h>
#include <hip/hip_bf16.h>

typedef __attribute__((ext_vector_type(2))) float v2f;
typedef __attribute__((ext_vector_type(8))) float v8f;

__device__ __forceinline__ v8f wmma_acc(v2f a, v2f b, v8f c) {
  return __builtin_amdgcn_wmma_f32_16x16x4_f32(
      /*neg_a=*/false, a, /*neg_b=*/false, b,
      /*c_mod=*/(short)0, c, /*reuse_a=*/false, /*reuse_b=*/false);
}

// ---------------------------------------------------------------------------
// Generic row-major matvec: out[r] = sum_c A[r, c] * x[c], A is [rows x cols].
// One 16-row tile per block; 8 waves split K (stride 32 columns/step, x4
// manual unroll => 8 b64 loads in flight per wave before 4 chained WMMAs).
// Uses V_WMMA_F32_16X16X4_F32 with B's 16 columns all equal to the x-chunk,
// so D's column 0 (lanes 0 and 16) carries the matvec result for 16 rows.
//   A 16x4 f32 layout: lane = half*16 + m -> A[m][half*2 + {0,1}]
//   B  4x16 f32 layout (symmetric): VGPR0 = K{0|2}, VGPR1 = K{1|3}
//   C/D 16x16 f32:  VGPR j, lane 0 -> D[j][0], lane 16 -> D[8+j][0]
// cols must be a multiple of 32 (true for 8192 / 4096 / 1024).
// ---------------------------------------------------------------------------
__global__ __launch_bounds__(256) void wmma_matvec_f32(
    const float* __restrict__ A, const float* __restrict__ x,
    float* __restrict__ out, int cols) {
  __shared__ float red[8][16];
  const int lane = threadIdx.x & 31;
  const int wave = threadIdx.x >> 5;
  const int half = lane >> 4;   // 0: lanes 0-15, 1: lanes 16-31
  const int mrow = lane & 15;   // row within the 16-row tile
  const size_t row0 = (size_t)blockIdx.x * 16;
  const int coff = wave * 4 + half * 2;  // this wave/half's first column
  const float* __restrict__ ap = A + (row0 + (size_t)mrow) * (size_t)cols + coff;
  const float* __restrict__ bp = x + coff;

  const int iters = cols >> 5;  // uniform (SGPR) trip count; %4==0 for our sizes
  v8f acc = {0.f, 0.f, 0.f, 0.f, 0.f, 0.f, 0.f, 0.f};

  int i = 0;
  for (; i + 4 <= iters; i += 4) {
    v2f a0 = *reinterpret_cast<const v2f*>(ap);
    v2f b0 = *reinterpret_cast<const v2f*>(bp);
    v2f a1 = *reinterpret_cast<const v2f*>(ap + 32);
    v2f b1 = *reinterpret_cast<const v2f*>(bp + 32);
    v2f a2 = *reinterpret_cast<const v2f*>(ap + 64);
    v2f b2 = *reinterpret_cast<const v2f*>(bp + 64);
    v2f a3 = *reinterpret_cast<const v2f*>(ap + 96);
    v2f b3 = *reinterpret_cast<const v2f*>(bp + 96);
    acc = wmma_acc(a0, b0, acc);
    acc = wmma_acc(a1, b1, acc);
    acc = wmma_acc(a2, b2, acc);
    acc = wmma_acc(a3, b3, acc);
    ap += 128;
    bp += 128;
  }
  for (; i < iters; ++i) {  // generic tail (never taken for 8192/4096/1024)
    v2f a0 = *reinterpret_cast<const v2f*>(ap);
    v2f b0 = *reinterpret_cast<const v2f*>(bp);
    acc = wmma_acc(a0, b0, acc);
    ap += 32;
    bp += 32;
  }

  // lanes 0 and 16 hold rows [0..7] and [8..15] of this wave's partial tile
  if (mrow == 0) {
#pragma unroll
    for (int j = 0; j < 8; ++j) red[wave][half * 8 + j] = acc[j];
  }
  __syncthreads();
  if (threadIdx.x < 16) {
    float s = 0.f;
#pragma unroll
    for (int w = 0; w < 8; ++w) s += red[w][threadIdx.x];
    out[row0 + threadIdx.x] = s;
  }
}

// ---------------------------------------------------------------------------
// Stage A: per-column partial stats over a D-slice (coalesced across columns):
//   p0 = sum s_in[d]*mem[d,m]   p1 = sum s_in[d]*keys[d,m]
//   p2 = sum s_q[d]*mem[d,m]    p3 = sum mem[d,m]^2
// ---------------------------------------------------------------------------
__global__ __launch_bounds__(256) void colstats_partial(
    const float* __restrict__ mem, const float* __restrict__ keys,
    const float* __restrict__ s_in, const float* __restrict__ s_q,
    float* __restrict__ partial, int M_, int rows_per) {
  const int m = blockIdx.x * 256 + threadIdx.x;
  const int d0 = blockIdx.y * rows_per;
  float p0 = 0.f, p1 = 0.f, p2 = 0.f, p3 = 0.f;
#pragma unroll 4
  for (int d = d0; d < d0 + rows_per; ++d) {
    const float si = s_in[d];                 // uniform -> scalar load
    const float sq = s_q[d];
    const float mv = mem[(size_t)d * M_ + m]; // coalesced over m
    const float kv = keys[(size_t)d * M_ + m];
    p0 += si * mv;
    p1 += si * kv;
    p2 += sq * mv;
    p3 += mv * mv;
  }
  float* base = partial + ((size_t)blockIdx.y * 4) * M_;
  base[0 * M_ + m] = p0;
  base[1 * M_ + m] = p1;
  base[2 * M_ + m] = p2;
  base[3 * M_ + m] = p3;
}

// Stage B: reduce splits, compute gate + combined normalize scale and logits.
//   g = sigmoid(d1+d2);  scale = (1+g)/max((1+g)*sqrt(ss), 1e-12)
//   logit = scale * d3   (== s_q . mem_hat[:,m])
__global__ __launch_bounds__(256) void colstats_finish(
    const float* __restrict__ partial, float* __restrict__ scale,
    float* __restrict__ logit, int M_, int splits) {
  const int m = blockIdx.x * 256 + threadIdx.x;
  float d1 = 0.f, d2 = 0.f, d3 = 0.f, ss = 0.f;
  for (int s = 0; s < splits; ++s) {
    const float* base = partial + ((size_t)s * 4) * M_;
    d1 += base[0 * M_ + m];
    d2 += base[1 * M_ + m];
    d3 += base[2 * M_ + m];
    ss += base[3 * M_ + m];
  }
  const float g = 1.f / (1.f + expf(-(d1 + d2)));
  const float sc = (1.f + g) / fmaxf((1.f + g) * sqrtf(ss), 1e-12f);
  scale[m] = sc;
  logit[m] = sc * d3;
}

// Softmax over M=1024 logits (single block), fused with q = p * scale so the
// following memory read-back is a plain row-major matvec u = mem @ q.
__global__ __launch_bounds__(1024) void softmax_q(
    const float* __restrict__ logit, const float* __restrict__ scale,
    float* __restrict__ q) {
  __shared__ float sdata[1024];
  const int t = threadIdx.x;
  const float v = logit[t];
  sdata[t] = v;
  __syncthreads();
  for (int s = 512; s > 0; s >>= 1) {
    if (t < s) sdata[t] = fmaxf(sdata[t], sdata[t + s]);
    __syncthreads();
  }
  const float mx = sdata[0];
  __syncthreads();
  const float e = expf(v - mx);
  sdata[t] = e;
  __syncthreads();
  for (int s = 512; s > 0; s >>= 1) {
    if (t < s) sdata[t] += sdata[t + s];
    __syncthreads();
  }
  q[t] = (e / sdata[0]) * scale[t];
}

// z = prelu(s_q + h, a_out)
__global__ __launch_bounds__(256) void add_prelu(
    const float* __restrict__ s_q, const float* __restrict__ h,
    const float* __restrict__ a, float* __restrict__ z, int n) {
  const int i = blockIdx.x * 256 + threadIdx.x;
  if (i < n) {
    const float v = s_q[i] + h[i];
    z[i] = v >= 0.f ? v : a[0] * v;
  }
}

extern "C" void kernel_launch(void* const* d_in, const int* in_sizes, int n_in,
                              void* d_out, int out_size, void* d_ws, size_t ws_size,
                              hipStream_t stream) {
  const float* input = (const float*)d_in[0];   // [1, L]
  const float* query = (const float*)d_in[1];   // [1, L]
  const float* F_i   = (const float*)d_in[2];   // [D, L]
  const float* F_q   = (const float*)d_in[3];   // [D, L]
  const float* keys  = (const float*)d_in[4];   // [D, M]
  const float* mem   = (const float*)d_in[5];   // [D, M]
  // d_in[6..8] = U, V, W : dead code in the reference forward (never used)
  const float* Rm    = (const float*)d_in[9];   // [D, D]
  const float* Hm    = (const float*)d_in[10];  // [D, D]
  // d_in[11] = a_mem : only feeds the dead candidate branch
  const float* a_out = (const float*)d_in[12];  // [1]

  const int L = in_sizes[0];
  const int D = in_sizes[2] / L;
  const int M = in_sizes[4] / D;
  const int SPLITS = 64;

  float* ws      = (float*)d_ws;
  float* s_in    = ws;               // D
  float* s_q     = s_in + D;         // D
  float* u       = s_q + D;          // D
  float* h       = u + D;            // D
  float* z       = h + D;            // D
  float* scale   = z + D;            // M
  float* logit   = scale + M;        // M
  float* q       = logit + M;        // M
  float* partial = q + M;            // SPLITS*4*M

  // 1-2: input/query projections (F @ v), 128 MB each — WMMA matvec path
  wmma_matvec_f32<<<D / 16, 256, 0, stream>>>(F_i, input, s_in, L);
  wmma_matvec_f32<<<D / 16, 256, 0, stream>>>(F_q, query, s_q, L);

  // 3-4: fused single pass over memory_nodes + keys -> gate, norm scale, logits
  colstats_partial<<<dim3(M / 256, SPLITS), 256, 0, stream>>>(
      mem, keys, s_in, s_q, partial, M, D / SPLITS);
  colstats_finish<<<M / 256, 256, 0, stream>>>(partial, scale, logit, M, SPLITS);

  // 5: attention weights fused with normalization scale: q = softmax(logit)*scale
  softmax_q<<<1, 1024, 0, stream>>>(logit, scale, q);

  // 6: u = mem @ q (row-major matvec, mem likely still resident in 192MB L2)
  wmma_matvec_f32<<<D / 16, 256, 0, stream>>>(mem, q, u, M);

  // 7-9: h = H@u ; z = prelu(s_q + h) ; y = R@z
  wmma_matvec_f32<<<D / 16, 256, 0, stream>>>(Hm, u, h, D);
  add_prelu<<<(D + 255) / 256, 256, 0, stream>>>(s_q, h, a_out, z, D);
  wmma_matvec_f32<<<D / 16, 256, 0, stream>>>(Rm, z, (float*)d_out, D);
}